// AttentionOperator_6167573037260
// MI455X (gfx1250) — compile-verified
//
#include <hip/hip_runtime.h>

// Chunked delta-rule recurrence for MI455X (gfx1250, wave32).
//  B=4,H=16,S=4096,D=64, chunk=64  -> 4096 chunks, one global sequential scan.
//  Pass 1 (parallel): A_i = I - k^T(t kb), b_i = k^T(t vb)        -> workspace
//  Pass 2 (1 WG):     state_{i+1} = A_i state_i + b_i  (serial scan), with
//                     GLOBAL_LOAD_ASYNC_TO_LDS_B128 double-buffering of A/b
//                     (ASYNCcnt) so HBM latency overlaps the WMMA matmul.
//  Pass 3 (parallel): u = t vb - w state_i ; o = q state_i + tril(q k^T) u
//  All 64x64x64 matmuls on V_WMMA_F32_16X16X4_F32 (fp32, matches reference).
//  Workspace: A | b | states = 3 * 64 MB = 192 MB of d_ws.

typedef __attribute__((ext_vector_type(2))) float v2f;
typedef __attribute__((ext_vector_type(8))) float v8f;

#define DDIM 64
#define CHNK 64
#define LS 68            // LDS row stride (floats): 16B aligned, bank-conflict free
#define NCHUNK 4096
#define TILE_FLOATS (64 * LS)

// --- one 16x16 tile of  acc += opA(A)[64x64] * opB(B)[64x64], K=64 ------------
template <bool AT, bool BT>
__device__ __forceinline__ v8f mm_tile(const float* __restrict__ A,
                                       const float* __restrict__ B,
                                       int mt, int nt, int lane, v8f acc) {
    const int mn = lane & 15;   // M for A-frag, N for B-frag (both lane halves)
    const int hi = lane >> 4;   // selects K pair {0,1} vs {2,3} within a K-step
#pragma unroll
    for (int kk = 0; kk < 16; ++kk) {
        const int k0 = kk * 4 + 2 * hi;
        v2f a, b;
        if (!AT) {
            a.x = A[(mt * 16 + mn) * LS + k0];
            a.y = A[(mt * 16 + mn) * LS + k0 + 1];
        } else {
            a.x = A[(k0) * LS + mt * 16 + mn];
            a.y = A[(k0 + 1) * LS + mt * 16 + mn];
        }
        if (!BT) {
            b.x = B[(k0) * LS + nt * 16 + mn];
            b.y = B[(k0 + 1) * LS + nt * 16 + mn];
        } else {
            b.x = B[(nt * 16 + mn) * LS + k0];
            b.y = B[(nt * 16 + mn) * LS + k0 + 1];
        }
        acc = __builtin_amdgcn_wmma_f32_16x16x4_f32(false, a, false, b,
                                                    (short)0, acc, false, false);
    }
    return acc;
}

__device__ __forceinline__ v8f load_tile(const float* __restrict__ S,
                                         int mt, int nt, int lane) {
    v8f c;
    const int mn = lane & 15, hi = lane >> 4;
#pragma unroll
    for (int r = 0; r < 8; ++r)
        c[r] = S[(mt * 16 + r + 8 * hi) * LS + nt * 16 + mn];
    return c;
}

__device__ __forceinline__ void store_tile(float* __restrict__ Dst,
                                           int mt, int nt, int lane, v8f c) {
    const int mn = lane & 15, hi = lane >> 4;
#pragma unroll
    for (int r = 0; r < 8; ++r)
        Dst[(mt * 16 + r + 8 * hi) * LS + nt * 16 + mn] = c[r];
}

// synchronous stage: contiguous 64x64 fp32 block -> LDS (stride LS)
__device__ __forceinline__ void stage(const float* __restrict__ g,
                                      float* __restrict__ s, int tid) {
#pragma unroll
    for (int i = 0; i < 4; ++i) {
        const int e = i * 1024 + tid * 4;
        const float4 v4 = *(const float4*)(g + e);
        *(float4*)(s + (e >> 6) * LS + (e & 63)) = v4;
    }
}

// async stage: GLOBAL_LOAD_ASYNC_TO_LDS_B128 (GV mode), tracked by ASYNCcnt.
// Per-lane LDS byte address = low 32 bits of the generic shared pointer.
__device__ __forceinline__ void stage_async(const float* __restrict__ g,
                                            float* s, int tid) {
    const uint32_t lds0 = (uint32_t)(uintptr_t)s;
#pragma unroll
    for (int i = 0; i < 4; ++i) {
        const int e = i * 1024 + tid * 4;
        const uint32_t loff =
            lds0 + (uint32_t)((((e >> 6) * LS) + (e & 63)) * 4);
        asm volatile("global_load_async_to_lds_b128 %0, %1, off"
                     :: "v"(loff), "v"(g + e)
                     : "memory");
    }
}

__device__ __forceinline__ void wait_async0() {
    asm volatile("s_wait_asynccnt 0x0" ::: "memory");
}

// =============================== Pass 1 ======================================
__global__ __launch_bounds__(256) void delta_pass1(
    const float* __restrict__ k, const float* __restrict__ v,
    const float* __restrict__ beta, float* __restrict__ Aout,
    float* __restrict__ bout) {
    __shared__ float k_s[TILE_FLOATS], kb_s[TILE_FLOATS], vb_s[TILE_FLOATS];
    __shared__ float t_s[TILE_FLOATS], w_s[TILE_FLOATS], u0_s[TILE_FLOATS];

    const int tid = threadIdx.x, lane = tid & 31, wid = tid >> 5;
    const int mn = lane & 15, hi = lane >> 4;
    const size_t base = (size_t)blockIdx.x * (CHNK * DDIM);

#pragma unroll
    for (int i = 0; i < 4; ++i) {
        const int e = i * 1024 + tid * 4;
        const float4 kv = *(const float4*)(k + base + e);
        const float4 bv = *(const float4*)(beta + base + e);
        const float4 vv = *(const float4*)(v + base + e);
        const int off = (e >> 6) * LS + (e & 63);
        *(float4*)(k_s + off) = kv;
        *(float4*)(kb_s + off) = float4{kv.x * bv.x, kv.y * bv.y, kv.z * bv.z, kv.w * bv.w};
        *(float4*)(vb_s + off) = float4{vv.x * bv.x, vv.y * bv.y, vv.z * bv.z, vv.w * bv.w};
    }
    __syncthreads();

    // t = I - strict_tril(kb @ k^T)
#pragma unroll
    for (int t2 = 0; t2 < 2; ++t2) {
        const int ti = wid * 2 + t2, mt = ti >> 2, nt = ti & 3;
        v8f acc = {};
        acc = mm_tile<false, true>(kb_s, k_s, mt, nt, lane, acc);
#pragma unroll
        for (int r = 0; r < 8; ++r) {
            const int row = mt * 16 + r + 8 * hi, col = nt * 16 + mn;
            t_s[row * LS + col] =
                (row == col ? 1.0f : 0.0f) - (row > col ? acc[r] : 0.0f);
        }
    }
    __syncthreads();

    // w = t @ kb ; u0 = t @ vb
#pragma unroll
    for (int t2 = 0; t2 < 2; ++t2) {
        const int ti = wid * 2 + t2, mt = ti >> 2, nt = ti & 3;
        v8f aw = {}, au = {};
        aw = mm_tile<false, false>(t_s, kb_s, mt, nt, lane, aw);
        au = mm_tile<false, false>(t_s, vb_s, mt, nt, lane, au);
        store_tile(w_s, mt, nt, lane, aw);
        store_tile(u0_s, mt, nt, lane, au);
    }
    __syncthreads();

    // A = I - k^T @ w ; b = k^T @ u0  -> workspace
#pragma unroll
    for (int t2 = 0; t2 < 2; ++t2) {
        const int ti = wid * 2 + t2, mt = ti >> 2, nt = ti & 3;
        v8f aA = {}, aB = {};
        aA = mm_tile<true, false>(k_s, w_s, mt, nt, lane, aA);
        aB = mm_tile<true, false>(k_s, u0_s, mt, nt, lane, aB);
#pragma unroll
        for (int r = 0; r < 8; ++r) {
            const int row = mt * 16 + r + 8 * hi, col = nt * 16 + mn;
            Aout[base + row * DDIM + col] = (row == col ? 1.0f : 0.0f) - aA[r];
            bout[base + row * DDIM + col] = aB[r];
        }
    }
}

// =============================== Pass 2 ======================================
// Single-WG serial scan with async double-buffering of A/b.
__global__ __launch_bounds__(256) void delta_pass2(
    const float* __restrict__ Aws, const float* __restrict__ bws,
    float* __restrict__ states, float* __restrict__ final_state) {
    __shared__ float A_s[2][TILE_FLOATS], b_s[2][TILE_FLOATS];
    __shared__ float st_s[2][TILE_FLOATS];

    const int tid = threadIdx.x, lane = tid & 31, wid = tid >> 5;
    for (int i = tid; i < TILE_FLOATS; i += 256) st_s[0][i] = 0.0f;

    // prologue: async-stage chunk 0 into buffer 0
    stage_async(Aws, &A_s[0][0], tid);
    stage_async(bws, &b_s[0][0], tid);

    int cur = 0;
    for (int c = 0; c < NCHUNK; ++c) {
        const size_t off = (size_t)c * (CHNK * DDIM);
        const int pb = c & 1;

        wait_async0();      // chunk c's A/b landed in LDS (this wave's DMAs)
        __syncthreads();    // visible to all waves; prev iter's readers done

        if (c + 1 < NCHUNK) {  // overlap next chunk's DMA with this matmul
            stage_async(Aws + off + CHNK * DDIM, &A_s[pb ^ 1][0], tid);
            stage_async(bws + off + CHNK * DDIM, &b_s[pb ^ 1][0], tid);
        }

        // spill incoming state for pass 3
#pragma unroll
        for (int i = 0; i < 4; ++i) {
            const int e = i * 1024 + tid * 4;
            *(float4*)(states + off + e) =
                *(const float4*)(&st_s[cur][(e >> 6) * LS + (e & 63)]);
        }

        // state' = A @ state + b
#pragma unroll
        for (int t2 = 0; t2 < 2; ++t2) {
            const int ti = wid * 2 + t2, mt = ti >> 2, nt = ti & 3;
            v8f acc = load_tile(&b_s[pb][0], mt, nt, lane);
            acc = mm_tile<false, false>(&A_s[pb][0], &st_s[cur][0], mt, nt,
                                        lane, acc);
            store_tile(&st_s[cur ^ 1][0], mt, nt, lane, acc);
        }
        __syncthreads();
        cur ^= 1;
    }
#pragma unroll
    for (int i = 0; i < 4; ++i) {
        const int e = i * 1024 + tid * 4;
        *(float4*)(final_state + e) =
            *(const float4*)(&st_s[cur][(e >> 6) * LS + (e & 63)]);
    }
}

// =============================== Pass 3 ======================================
__global__ __launch_bounds__(256) void delta_pass3(
    const float* __restrict__ q, const float* __restrict__ k,
    const float* __restrict__ v, const float* __restrict__ beta,
    const float* __restrict__ states, float* __restrict__ out) {
    __shared__ float q_s[TILE_FLOATS], k_s[TILE_FLOATS], kb_s[TILE_FLOATS];
    __shared__ float vb_s[TILE_FLOATS], t_s[TILE_FLOATS], w_s[TILE_FLOATS];
    __shared__ float u_s[TILE_FLOATS], st_s[TILE_FLOATS];

    const int tid = threadIdx.x, lane = tid & 31, wid = tid >> 5;
    const int mn = lane & 15, hi = lane >> 4;
    const size_t base = (size_t)blockIdx.x * (CHNK * DDIM);

#pragma unroll
    for (int i = 0; i < 4; ++i) {
        const int e = i * 1024 + tid * 4;
        const float4 qv = *(const float4*)(q + base + e);
        const float4 kv = *(const float4*)(k + base + e);
        const float4 bv = *(const float4*)(beta + base + e);
        const float4 vv = *(const float4*)(v + base + e);
        const float4 sv = *(const float4*)(states + base + e);
        const int off = (e >> 6) * LS + (e & 63);
        *(float4*)(q_s + off) = qv;
        *(float4*)(k_s + off) = kv;
        *(float4*)(kb_s + off) = float4{kv.x * bv.x, kv.y * bv.y, kv.z * bv.z, kv.w * bv.w};
        *(float4*)(vb_s + off) = float4{vv.x * bv.x, vv.y * bv.y, vv.z * bv.z, vv.w * bv.w};
        *(float4*)(st_s + off) = sv;
    }
    __syncthreads();

    // t = I - strict_tril(kb @ k^T)
#pragma unroll
    for (int t2 = 0; t2 < 2; ++t2) {
        const int ti = wid * 2 + t2, mt = ti >> 2, nt = ti & 3;
        v8f acc = {};
        acc = mm_tile<false, true>(kb_s, k_s, mt, nt, lane, acc);
#pragma unroll
        for (int r = 0; r < 8; ++r) {
            const int row = mt * 16 + r + 8 * hi, col = nt * 16 + mn;
            t_s[row * LS + col] =
                (row == col ? 1.0f : 0.0f) - (row > col ? acc[r] : 0.0f);
        }
    }
    __syncthreads();

    // w = t @ kb ; u0 = t @ vb
#pragma unroll
    for (int t2 = 0; t2 < 2; ++t2) {
        const int ti = wid * 2 + t2, mt = ti >> 2, nt = ti & 3;
        v8f aw = {}, au = {};
        aw = mm_tile<false, false>(t_s, kb_s, mt, nt, lane, aw);
        au = mm_tile<false, false>(t_s, vb_s, mt, nt, lane, au);
        store_tile(w_s, mt, nt, lane, aw);
        store_tile(u_s, mt, nt, lane, au);
    }
    __syncthreads();

    // u -= w @ state
#pragma unroll
    for (int t2 = 0; t2 < 2; ++t2) {
        const int ti = wid * 2 + t2, mt = ti >> 2, nt = ti & 3;
        v8f acc = {};
        acc = mm_tile<false, false>(w_s, st_s, mt, nt, lane, acc);
#pragma unroll
        for (int r = 0; r < 8; ++r) {
            const int row = mt * 16 + r + 8 * hi, col = nt * 16 + mn;
            u_s[row * LS + col] -= acc[r];
        }
    }
    __syncthreads();

    // S = tril(q @ k^T) -> reuse t_s
#pragma unroll
    for (int t2 = 0; t2 < 2; ++t2) {
        const int ti = wid * 2 + t2, mt = ti >> 2, nt = ti & 3;
        v8f acc = {};
        acc = mm_tile<false, true>(q_s, k_s, mt, nt, lane, acc);
#pragma unroll
        for (int r = 0; r < 8; ++r) {
            const int row = mt * 16 + r + 8 * hi, col = nt * 16 + mn;
            t_s[row * LS + col] = (row >= col) ? acc[r] : 0.0f;
        }
    }
    __syncthreads();

    // o = q @ state + S @ u  -> global
#pragma unroll
    for (int t2 = 0; t2 < 2; ++t2) {
        const int ti = wid * 2 + t2, mt = ti >> 2, nt = ti & 3;
        v8f acc = {};
        acc = mm_tile<false, false>(q_s, st_s, mt, nt, lane, acc);
        acc = mm_tile<false, false>(t_s, u_s, mt, nt, lane, acc);
#pragma unroll
        for (int r = 0; r < 8; ++r) {
            const int row = mt * 16 + r + 8 * hi, col = nt * 16 + mn;
            out[base + row * DDIM + col] = acc[r];
        }
    }
}

// =============================================================================
extern "C" void kernel_launch(void* const* d_in, const int* in_sizes, int n_in,
                              void* d_out, int out_size, void* d_ws,
                              size_t ws_size, hipStream_t stream) {
    (void)in_sizes; (void)n_in; (void)out_size; (void)ws_size;
    const float* q    = (const float*)d_in[0];
    const float* k    = (const float*)d_in[1];
    const float* v    = (const float*)d_in[2];
    const float* beta = (const float*)d_in[3];
    // d_in[4] = chunk_size scalar (fixed at 64 for this problem instance)

    float* out = (float*)d_out;                                // [262144, 64]
    float* final_state = out + (size_t)NCHUNK * CHNK * DDIM;   // [64, 64]

    const size_t blk = (size_t)NCHUNK * CHNK * DDIM;           // 16M floats
    float* Aws    = (float*)d_ws;
    float* bws    = Aws + blk;
    float* states = bws + blk;

    delta_pass1<<<NCHUNK, 256, 0, stream>>>(k, v, beta, Aws, bws);
    delta_pass2<<<1, 256, 0, stream>>>(Aws, bws, states, final_state);
    delta_pass3<<<NCHUNK, 256, 0, stream>>>(q, k, v, beta, states, out);
}